// ROIAlign_8993661518501
// MI455X (gfx1250) — compile-verified
//
#include <hip/hip_runtime.h>
#include <stdint.h>

// ---------------------------------------------------------------------------
// FPN ROIAlign for MI455X (gfx1250), TDM-gather pipelined version.
//
// 1 workgroup per ROI (2048 blocks), 256 threads = 8 wave32, one thread per
// channel. Per output bin, the 16 (sample x corner) feature rows (each
// C=256 f32 = 1KB, contiguous in NHWC) are gathered by the Tensor Data Mover
// in gather mode: one tensor_load_to_lds moves all 16 rows into LDS.
//
// Pipeline: 3 LDS buffers, 2 TDM gathers in flight per block (TDM ops from
// one wave complete in order, so s_wait_tensorcnt(1) after issuing bin i+2
// guarantees bin i+1 has landed). ~6 blocks/WGP x 2 outstanding gathers keeps
// enough bytes in flight to run at HBM bandwidth instead of gather latency.
// ---------------------------------------------------------------------------

typedef unsigned int v4u __attribute__((ext_vector_type(4)));
typedef int          v8i __attribute__((ext_vector_type(8)));
typedef int          v4i __attribute__((ext_vector_type(4)));

#define CCH   256
#define SOUT  7
#define NBIN  49

__device__ __forceinline__ int imin(int a, int b) { return a < b ? a : b; }
__device__ __forceinline__ int imax(int a, int b) { return a > b ? a : b; }

__global__ __launch_bounds__(256, 4) void roialign_tdm_kernel(
    const float* __restrict__ fm2, const float* __restrict__ fm3,
    const float* __restrict__ fm4, const float* __restrict__ fm5,
    const float* __restrict__ rois, float* __restrict__ out, int nPerBatch)
{
    // Three 16-row staging buffers: 3 * 16 * 256 * 4B = 48 KB LDS.
    __shared__ float smem[3][16][CCH];

    const int rid = blockIdx.x;          // global ROI id, 0..B*N-1
    const int b   = rid / nPerBatch;     // batch index
    const int tid = threadIdx.x;         // channel index 0..255

    // --- ROI box (original image coords) ---
    const float x1 = rois[rid * 4 + 0];
    const float y1 = rois[rid * 4 + 1];
    const float x2 = rois[rid * 4 + 2];
    const float y2 = rois[rid * 4 + 3];

    // --- FPN level: clip(round(log2(sqrt(area)/224) + 4), 2, 5) ---
    const float area = (y2 - y1) * (x2 - x1);
    const float lf   = log2f(sqrtf(area) * (1.0f / 224.0f)) + 4.0f;
    int lvl = (int)rintf(lf);            // rintf = round-half-even, matches jnp.round
    lvl = imin(imax(lvl, 2), 5);

    const float* fm; int H;
    switch (lvl) {
        case 2:  fm = fm2; H = 256; break;
        case 3:  fm = fm3; H = 128; break;
        case 4:  fm = fm4; H = 64;  break;
        default: fm = fm5; H = 32;  break;
    }
    const int   W     = H;
    const float scale = 1.0f / (float)(4 << (lvl - 2));

    // --- ROI in feature-map coords ---
    const float bx1 = x1 * scale, by1 = y1 * scale;
    const float rw  = fmaxf(x2 * scale - bx1, 1.0f);
    const float rh  = fmaxf(y2 * scale - by1, 1.0f);
    const float wstep = rw * (1.0f / (float)SOUT);
    const float hstep = rh * (1.0f / (float)SOUT);

    const float* fmb  = fm + (size_t)b * H * W * CCH;   // batch's [H*W, C] tensor
    float*       outb = out + (size_t)rid * CCH * NBIN; // this ROI's [C, 7, 7]

    float w[16];   // bilinear weights (x0.25 mean, x valid-mask), block-uniform
    int   idx[16]; // gathered row indices (cy*W + cx), block-uniform

    // Per-bin sample/corner weights and row indices (all block-uniform; the
    // compiler scalarizes this whole section onto the SALU).
    auto prep = [&](int bin) {
        const int by = bin / SOUT, bx = bin % SOUT;
#pragma unroll
        for (int s = 0; s < 4; ++s) {
            const int sy = s >> 1, sx = s & 1;
            const float yy = by1 + ((float)(2 * by + sy) + 0.5f) * 0.5f * hstep;
            const float xx = bx1 + ((float)(2 * bx + sx) + 0.5f) * 0.5f * wstep;
            const bool vy = (yy > -1.0f) && (yy < (float)H);
            const bool vx = (xx > -1.0f) && (xx < (float)W);
            const float yc = fminf(fmaxf(yy, 0.0f), (float)(H - 1));
            const float xc = fminf(fmaxf(xx, 0.0f), (float)(W - 1));
            const float y0f = floorf(yc), x0f = floorf(xc);
            const int iy0 = (int)y0f, ix0 = (int)x0f;
            const int iy1 = imin(iy0 + 1, H - 1);
            const int ix1 = imin(ix0 + 1, W - 1);
            const float ly = yc - y0f, lx = xc - x0f;
            const float hy = 1.0f - ly, hx = 1.0f - lx;
            const float m  = (vy && vx) ? 0.25f : 0.0f;  // fold 2x2 mean into weight
            w[s * 4 + 0] = m * hy * hx;  idx[s * 4 + 0] = iy0 * W + ix0;
            w[s * 4 + 1] = m * hy * lx;  idx[s * 4 + 1] = iy0 * W + ix1;
            w[s * 4 + 2] = m * ly * hx;  idx[s * 4 + 2] = iy1 * W + ix0;
            w[s * 4 + 3] = m * ly * lx;  idx[s * 4 + 3] = iy1 * W + ix1;
        }
    };

    // TDM gather-mode descriptor (CDNA5 ISA ch.8): gather 16 rows of 256 f32
    // from the [H*W, 256] tensor into smem[buf]. Issued by wave 0 only;
    // TDM ignores EXEC and signals TENSORcnt once per instruction.
    auto issue = [&](int buf) {
        const uint32_t lds = (uint32_t)(uintptr_t)(&smem[buf][0][0]); // LDS byte offset
        const uint64_t ga  = (uint64_t)(uintptr_t)fmb;                // bytes
        const int HW = H * W;

        v4u g0;
        g0[0] = 0x80000001u;                 // count=1 | gather_index_size=16b | gather_mode=1
        g0[1] = lds;                         // lds_addr [63:32]
        g0[2] = (uint32_t)ga;                // global_addr lo
        g0[3] = (uint32_t)((ga >> 32) & 0x01FFFFFFu) | (2u << 30); // addr[56:32] | type=2

        v8i g1;
        g1[0] = (int)(2u << 16);                                  // data_size=4B, mask=0
        g1[1] = (int)((CCH & 0xFFFF) << 16);                      // tensor_dim0[15:0]
        g1[2] = (int)(((CCH >> 16) & 0xFFFF) | ((HW & 0xFFFF) << 16)); // dim0 hi | dim1 lo
        g1[3] = (int)(((HW >> 16) & 0xFFFF) | (CCH << 16));       // dim1 hi | tile_dim0=256
        g1[4] = 16;                                               // tile_dim1 = #indices
        g1[5] = CCH;                                              // tensor_dim0_stride
        g1[6] = 0;
        g1[7] = 0;

        v4i g2, g3;
#pragma unroll
        for (int k = 0; k < 4; ++k)
            g2[k] = (idx[2 * k] & 0xFFFF) | (idx[2 * k + 1] << 16);      // rows 0..7
#pragma unroll
        for (int k = 0; k < 4; ++k)
            g3[k] = (idx[8 + 2 * k] & 0xFFFF) | (idx[8 + 2 * k + 1] << 16); // rows 8..15

        // amdgpu-toolchain (clang-23) form: 6 args — groups 0..3 in ISA order,
        // a zero-filled trailing int32x8 (reserved descriptor space), cpol=0.
        v8i g4 = { 0, 0, 0, 0, 0, 0, 0, 0 };
        __builtin_amdgcn_tensor_load_to_lds(g0, g1, g2, g3, g4, 0);
    };

    // --- prologue: prime 2 gathers (bins 0 and 1) ---
    float wcur[16], wnext[16];
    prep(0);
    if (tid < 32) issue(0);
#pragma unroll
    for (int j = 0; j < 16; ++j) wcur[j] = w[j];
    prep(1);
    if (tid < 32) issue(1);
#pragma unroll
    for (int j = 0; j < 16; ++j) wnext[j] = w[j];

    __builtin_amdgcn_s_wait_tensorcnt(1);  // oldest (bin 0) has landed
    __syncthreads();

    for (int bin = 0; bin < NBIN; ++bin) {
        const int cur = bin % 3;

        // Kick off bin+2's DMA into the buffer last read at bin-1 (safe: that
        // read was fenced by the previous barrier).
        if (bin + 2 < NBIN) {
            prep(bin + 2);
            if (tid < 32) issue((bin + 2) % 3);
        }

        // out[c, by, bx] = sum_j w[j] * F[row_j, c]; conflict-free LDS reads.
        float acc = 0.0f;
#pragma unroll
        for (int j = 0; j < 16; ++j)
            acc = fmaf(wcur[j], smem[cur][j][tid], acc);
        outb[tid * NBIN + bin] = acc;

        // Rotate the 2-deep weight pipeline: wcur <- bin+1, wnext <- bin+2.
#pragma unroll
        for (int j = 0; j < 16; ++j) { wcur[j] = wnext[j]; wnext[j] = w[j]; }

        // Guarantee bin+1's buffer has landed before the next iteration.
        // In-order TDM completion: with bin+2 still in flight wait cnt<=1,
        // once nothing more was issued wait cnt<=0.
        if (bin + 2 < NBIN)
            __builtin_amdgcn_s_wait_tensorcnt(1);
        else
            __builtin_amdgcn_s_wait_tensorcnt(0);
        __syncthreads();
    }
}

extern "C" void kernel_launch(void* const* d_in, const int* in_sizes, int n_in,
                              void* d_out, int out_size, void* d_ws, size_t ws_size,
                              hipStream_t stream) {
    const float* fm2  = (const float*)d_in[0];
    const float* fm3  = (const float*)d_in[1];
    const float* fm4  = (const float*)d_in[2];
    const float* fm5  = (const float*)d_in[3];
    const float* rois = (const float*)d_in[4];
    float*       out  = (float*)d_out;

    const int totalR = in_sizes[4] / 4;                    // B*N rois
    const int B      = in_sizes[0] / (256 * 256 * CCH);    // from fm2 [B,256,256,256]
    const int nPerB  = totalR / B;

    dim3 grid(totalR), block(256);
    hipLaunchKernelGGL(roialign_tdm_kernel, grid, block, 0, stream,
                       fm2, fm3, fm4, fm5, rois, out, nPerB);
}